// DigitCapsLayer_14053132993108
// MI455X (gfx1250) — compile-verified
//
#include <hip/hip_runtime.h>
#include <math.h>

typedef float v2f __attribute__((ext_vector_type(2)));
typedef float v8f __attribute__((ext_vector_type(8)));

#define BATCH 512
#define NPRIM 1152   // 32 maps * 36 prim caps
#define NCLS  10
#define IDIM  8
#define JO    160    // NCLS * OUT_CAP_DIM

// --------------------------------------------------------------------------
// Phase 1: u_hat[b,n,jo] = sum_i x[b,n,i] * W[n,jo,i]
// Per n: [512 x 8] @ [8 x 160] GEMM, done as 16x16 tiles with two
// V_WMMA_F32_16X16X4_F32 (K = 0..3 and 4..7). One wave = one (n, b-tile),
// A operand reused across all 10 jo-tiles.
//
// 16x4 f32 A layout: lanes 0-15 = rows, VGPR0/1 = K0/K1; lanes 16-31 = K2/K3.
// B (4x16, K x N) mirrors this with N across lanes. C/D: VGPR r -> row r
// (lanes 0-15) / row r+8 (lanes 16-31), col = lane&15.
// --------------------------------------------------------------------------
__global__ __launch_bounds__(256) void uhat_wmma_kernel(
    const float* __restrict__ x,   // [B][NPRIM][8]
    const float* __restrict__ W,   // [NPRIM][160][8]
    float* __restrict__ uh)        // [B][NPRIM][160]
{
  const int lane = threadIdx.x & 31;
  const int wid  = blockIdx.x * (blockDim.x >> 5) + (threadIdx.x >> 5);
  const int n    = wid >> 5;          // 1152 values
  const int btl  = wid & 31;          // 32 b-tiles of 16
  if (n >= NPRIM) return;

  const int b0  = btl * 16;
  const int kh  = lane >> 4;          // 0 -> K{0,1}/{4,5}; 1 -> K{2,3}/{6,7}
  const int r15 = lane & 15;

  const float* xr = x + ((size_t)(b0 + r15) * NPRIM + n) * IDIM;
  v2f a0 = *(const v2f*)(xr + 2 * kh);       // K = 2kh, 2kh+1
  v2f a1 = *(const v2f*)(xr + 4 + 2 * kh);   // K = 4+2kh, 5+2kh

  const float* Wn = W + (size_t)n * JO * IDIM;

  for (int t = 0; t < 10; ++t) {             // jo tiles of 16
    const float* wr = Wn + (t * 16 + r15) * IDIM;
    v2f bb0 = *(const v2f*)(wr + 2 * kh);
    v2f bb1 = *(const v2f*)(wr + 4 + 2 * kh);

    v8f acc = {0.f, 0.f, 0.f, 0.f, 0.f, 0.f, 0.f, 0.f};
    acc = __builtin_amdgcn_wmma_f32_16x16x4_f32(false, a0, false, bb0,
                                                (short)0, acc, false, false);
    acc = __builtin_amdgcn_wmma_f32_16x16x4_f32(false, a1, false, bb1,
                                                (short)0, acc, false, false);

    const size_t base =
        ((size_t)(b0 + 8 * kh) * NPRIM + n) * JO + t * 16 + r15;
#pragma unroll
    for (int r = 0; r < 8; ++r)
      uh[base + (size_t)r * NPRIM * JO] = acc[r];
  }
}

// --------------------------------------------------------------------------
// Phase 2: dynamic routing, one block per batch element, all iterations
// fused. b_ij lives in LDS (45 KB). Per iteration a single streaming pass
// over uh[b]: agreement (iter>0) -> b update -> e=exp(b) -> accumulate
// s_unnorm and Z, then v = squash(s_unnorm / Z).
//
// 256 threads = 64 groups of 4 lanes; group g handles n = step*64+g,
// lane q owns o-quad [4q..4q+3] (float4 loads).
// --------------------------------------------------------------------------
__global__ __launch_bounds__(256) void routing_kernel(
    const float* __restrict__ uh,   // [B][NPRIM][160]
    float* __restrict__ out,        // [B][1][10][16][1] flat = b*160 + j*16 + o
    const int* __restrict__ n_iter_p)
{
  __shared__ float bij[NPRIM * NCLS];          // 46080 B
  __shared__ float red[8 * 4 * NCLS * 4];      // [wave][q][j][4]
  __shared__ float zred[8 * NCLS];             // [wave][j]
  __shared__ float vsh[JO];

  const int tid  = threadIdx.x;
  const int b    = blockIdx.x;
  const int lane = tid & 31;
  const int wv   = tid >> 5;
  const int q    = lane & 3;      // o-quad
  const int gw   = lane >> 2;     // group within wave (0..7)
  const int g    = wv * 8 + gw;   // group 0..63

  const int NI = *n_iter_p;
  const float* uhb = uh + (size_t)b * NPRIM * JO;

  for (int i = tid; i < NPRIM * NCLS; i += 256) bij[i] = 0.f;
  __syncthreads();

  for (int it = 0; it < NI; ++it) {
    float sacc[NCLS][4];
    float zacc[NCLS];
#pragma unroll
    for (int j = 0; j < NCLS; ++j) {
      zacc[j] = 0.f;
      sacc[j][0] = sacc[j][1] = sacc[j][2] = sacc[j][3] = 0.f;
    }

    for (int step = 0; step < NPRIM / 64; ++step) {   // 18 steps
      const int n = step * 64 + g;
      const float* ur = uhb + (size_t)n * JO + q * 4;
      __builtin_prefetch(ur + 64 * JO, 0, 0);         // next tile -> global_prefetch_b8

      float4 u[NCLS];
#pragma unroll
      for (int j = 0; j < NCLS; ++j)
        u[j] = *(const float4*)(ur + j * 16);

      float e[NCLS];
      if (it > 0) {
#pragma unroll
        for (int j = 0; j < NCLS; ++j) {
          const float* vj = &vsh[j * 16 + q * 4];
          float p = u[j].x * vj[0] + u[j].y * vj[1] +
                    u[j].z * vj[2] + u[j].w * vj[3];
          p += __shfl_xor(p, 1, 32);       // reduce over the 4-lane quad
          p += __shfl_xor(p, 2, 32);
          float bn = bij[n * NCLS + j] + p;
          if (q == 0) bij[n * NCLS + j] = bn;
          e[j] = __expf(bn);
        }
      } else {
#pragma unroll
        for (int j = 0; j < NCLS; ++j) e[j] = 1.0f;   // exp(0), b starts at 0
      }

#pragma unroll
      for (int j = 0; j < NCLS; ++j) {
        zacc[j]    += e[j];
        sacc[j][0] += e[j] * u[j].x;
        sacc[j][1] += e[j] * u[j].y;
        sacc[j][2] += e[j] * u[j].z;
        sacc[j][3] += e[j] * u[j].w;
      }
    }

    // Reduce over the 8 groups of this wave: combine lanes with equal q.
#pragma unroll
    for (int j = 0; j < NCLS; ++j) {
#pragma unroll
      for (int c = 0; c < 4; ++c) {
        sacc[j][c] += __shfl_xor(sacc[j][c], 4, 32);
        sacc[j][c] += __shfl_xor(sacc[j][c], 8, 32);
        sacc[j][c] += __shfl_xor(sacc[j][c], 16, 32);
      }
      zacc[j] += __shfl_xor(zacc[j], 4, 32);
      zacc[j] += __shfl_xor(zacc[j], 8, 32);
      zacc[j] += __shfl_xor(zacc[j], 16, 32);
    }
    if (gw == 0) {                      // lanes 0..3 hold wave totals, q = lane
#pragma unroll
      for (int j = 0; j < NCLS; ++j) {
#pragma unroll
        for (int c = 0; c < 4; ++c)
          red[((wv * 4 + q) * NCLS + j) * 4 + c] = sacc[j][c];
      }
      if (q == 0)
#pragma unroll
        for (int j = 0; j < NCLS; ++j) zred[wv * NCLS + j] = zacc[j];
    }
    __syncthreads();

    // 160 threads: (j,o) -> softmax normalization + squash.
    if (tid < JO) {
      const int j  = tid >> 4;
      const int o  = tid & 15;
      const int qq = o >> 2;
      const int oi = o & 3;
      float s = 0.f, Z = 0.f;
#pragma unroll
      for (int w = 0; w < 8; ++w) {
        s += red[((w * 4 + qq) * NCLS + j) * 4 + oi];
        Z += zred[w * NCLS + j];
      }
      s /= Z;                                  // s_j[o]
      float sq = s * s;                        // |s|^2 over o (16 lanes)
      sq += __shfl_xor(sq, 1, 32);
      sq += __shfl_xor(sq, 2, 32);
      sq += __shfl_xor(sq, 4, 32);
      sq += __shfl_xor(sq, 8, 32);
      float v = s * (sq / ((1.0f + sq) * sqrtf(sq + 1e-9f)));
      vsh[tid] = v;
      out[(size_t)b * JO + tid] = v;           // last iteration's write wins
    }
    __syncthreads();
  }
}

extern "C" void kernel_launch(void* const* d_in, const int* in_sizes, int n_in,
                              void* d_out, int out_size, void* d_ws, size_t ws_size,
                              hipStream_t stream) {
  const float* x  = (const float*)d_in[0];   // [512][1152][8]
  const float* W  = (const float*)d_in[1];   // [1152][160][8]
  const int* n_it = (const int*)d_in[2];     // scalar (==3) in device mem
  float* uhat = (float*)d_ws;                // 512*1152*160 f32 = 377,487,360 B

  // 1152 n * 32 b-tiles = 36864 wave tasks, 8 waves per 256-thread block
  uhat_wmma_kernel<<<4608, 256, 0, stream>>>(x, W, uhat);
  routing_kernel<<<BATCH, 256, 0, stream>>>(uhat, (float*)d_out, n_it);
}